// SparseLinear_7619271983253
// MI455X (gfx1250) — compile-verified
//
#include <hip/hip_runtime.h>

typedef __attribute__((ext_vector_type(2))) float v2f;
typedef __attribute__((ext_vector_type(8))) float v8f;

#define M_TOT 16384   // B * S = 8 * 2048
#define N_TOT 4096    // D_OUT
#define K_TOT 4096    // D_IN

#define BK      32    // K-step (128 bytes per row)
#define LDS_STR 36    // padded row stride in floats -> bank-conflict-free ds_load_b64
#define KSTEPS  (K_TOT / BK)   // 128

// ---- CDNA5 async Global->LDS copy (ASYNCcnt-tracked, no VGPR round trip) ----
__device__ __forceinline__ void async_copy_b128(const float* gaddr, unsigned lds_off)
{
    // VDST = VGPR holding LDS byte address, VADDR = 64-bit global address
    asm volatile("global_load_async_to_lds_b128 %0, %1, off"
                 :: "v"(lds_off), "v"(gaddr) : "memory");
}

__device__ __forceinline__ void wait_async_zero()
{
    asm volatile("s_wait_asynccnt 0" ::: "memory");
}

// Block: 256 threads = 8 wave32.  Block tile: 128(M) x 64(N), wave tile 16x64.
// Double-buffered LDS staging; compute via V_WMMA_F32_16X16X4_F32.
__global__ __launch_bounds__(256)
void sparse_linear_wmma_f32(const float* __restrict__ x,
                            const float* __restrict__ w,
                            const float* __restrict__ bias,
                            float* __restrict__ out)
{
    __shared__ float ldsX[2][128 * LDS_STR];   // 2 x 18 KB
    __shared__ float ldsW[2][ 64 * LDS_STR];   // 2 x  9 KB

    const int tid    = threadIdx.x;
    const int lane   = tid & 31;
    const int waveId = tid >> 5;    // 0..7
    const int half   = lane >> 4;   // 0 or 1
    const int lm     = lane & 15;   // row (A) / col (B,C)

    const int m0 = blockIdx.y * 128;           // block M base
    const int n0 = blockIdx.x * 64;            // block N base
    const int mw = waveId * 16;                // wave M offset within block

    // Cooperative tile-load geometry: 8 threads per 32-float row (b128 each).
    const int lr = tid >> 3;        // 0..31 : row within 32-row group
    const int lc = (tid & 7) * 4;   // 0,4,...,28 : float column

    // Issue all async loads for K-step k0 into LDS buffer `buf`.
    auto issue_tile = [&](int k0, int buf) {
        const float* gx = x + (size_t)(m0 + lr) * K_TOT + k0 + lc;
        const float* gw = w + (size_t)(n0 + lr) * K_TOT + k0 + lc;
        float* lx = &ldsX[buf][lr * LDS_STR + lc];
        float* lw = &ldsW[buf][lr * LDS_STR + lc];
#pragma unroll
        for (int i = 0; i < 4; ++i)   // 4 x 32 = 128 x rows
            async_copy_b128(gx + (size_t)(i * 32) * K_TOT,
                            (unsigned)(uintptr_t)(lx + i * 32 * LDS_STR));
#pragma unroll
        for (int i = 0; i < 2; ++i)   // 2 x 32 = 64 w rows
            async_copy_b128(gw + (size_t)(i * 32) * K_TOT,
                            (unsigned)(uintptr_t)(lw + i * 32 * LDS_STR));
    };

    v8f acc0 = {}, acc1 = {}, acc2 = {}, acc3 = {};

    issue_tile(0, 0);

    for (int step = 0; step < KSTEPS; ++step) {
        const int buf = step & 1;
        wait_async_zero();     // this wave's async LDS writes complete
        __syncthreads();       // all waves' tile data visible; prev buffer free

        if (step + 1 < KSTEPS)
            issue_tile((step + 1) * BK, buf ^ 1);

        const float* xs = &ldsX[buf][(mw + lm) * LDS_STR + 2 * half];
        const float* ws = &ldsW[buf][lm * LDS_STR + 2 * half];

#pragma unroll
        for (int kk = 0; kk < BK / 4; ++kk) {   // 8 sub-steps of K=4
            const int ko = kk * 4;
            v2f a  = *(const v2f*)(xs + ko);
            v2f b0 = *(const v2f*)(ws + ko);
            v2f b1 = *(const v2f*)(ws + 16 * LDS_STR + ko);
            v2f b2 = *(const v2f*)(ws + 32 * LDS_STR + ko);
            v2f b3 = *(const v2f*)(ws + 48 * LDS_STR + ko);
            acc0 = __builtin_amdgcn_wmma_f32_16x16x4_f32(false, a, false, b0,
                                                         (short)0, acc0, false, false);
            acc1 = __builtin_amdgcn_wmma_f32_16x16x4_f32(false, a, false, b1,
                                                         (short)0, acc1, false, false);
            acc2 = __builtin_amdgcn_wmma_f32_16x16x4_f32(false, a, false, b2,
                                                         (short)0, acc2, false, false);
            acc3 = __builtin_amdgcn_wmma_f32_16x16x4_f32(false, a, false, b3,
                                                         (short)0, acc3, false, false);
        }
        __syncthreads();       // all waves done with `buf` before it is reloaded
    }

    const float bv0 = bias[n0 + lm];
    const float bv1 = bias[n0 + 16 + lm];
    const float bv2 = bias[n0 + 32 + lm];
    const float bv3 = bias[n0 + 48 + lm];

    float* orow = out + (size_t)(m0 + mw) * N_TOT + n0 + lm;
#pragma unroll
    for (int r = 0; r < 8; ++r) {
        const size_t off = (size_t)(r + half * 8) * N_TOT;
        orow[off +  0] = acc0[r] + bv0;
        orow[off + 16] = acc1[r] + bv1;
        orow[off + 32] = acc2[r] + bv2;
        orow[off + 48] = acc3[r] + bv3;
    }
}

extern "C" void kernel_launch(void* const* d_in, const int* in_sizes, int n_in,
                              void* d_out, int out_size, void* d_ws, size_t ws_size,
                              hipStream_t stream)
{
    const float* x  = (const float*)d_in[0];   // (8, 2048, 4096) fp32
    const float* w  = (const float*)d_in[1];   // (4096, 4096) fp32
    const float* b  = (const float*)d_in[2];   // (4096,) fp32
    float* out = (float*)d_out;                // (8, 2048, 4096) fp32

    dim3 grid(N_TOT / 64, M_TOT / 128);        // (64, 128)
    dim3 block(256);
    hipLaunchKernelGGL(sparse_linear_wmma_f32, grid, block, 0, stream, x, w, b, out);
}